// ScaleDotProductAttention_81123342286995
// MI455X (gfx1250) — compile-verified
//
#include <hip/hip_runtime.h>
#include <hip/hip_bf16.h>
#include <math.h>

typedef float v2f __attribute__((ext_vector_type(2)));
typedef float v4f __attribute__((ext_vector_type(4)));
typedef float v8f __attribute__((ext_vector_type(8)));

#define B_  2
#define H_  16
#define S_  2048
#define D_  64
#define QROWS 16
#define NWAVES 8

// 10000^(-p/32) = exp2(-p * log2(10000)/32);  log2(10000)/32 = 0.415241011860920...
__device__ __forceinline__ float inv_freq(int p) {
    return exp2f(-0.41524101186092030f * (float)p);
}

extern "C" __global__ void __launch_bounds__(256, 1)
rope_attn_fused(const float* __restrict__ q, const float* __restrict__ k,
                const float* __restrict__ v, const int* __restrict__ mask,
                float* __restrict__ out, float* __restrict__ score)
{
    __shared__ float q_lds[QROWS * 68];          // padded: stride 68 floats -> conflict-free frag reads
    __shared__ float redmax[NWAVES][16];
    __shared__ float redsum[NWAVES][16];
    __shared__ float out_acc[QROWS * 66];        // padded: stride 66 -> conflict-free ds_add_f32

    const int tid  = threadIdx.x;
    const int w    = tid >> 5;                   // wave id 0..7
    const int lane = tid & 31;
    const int r    = lane & 15;                  // row/col within 16
    const int hi   = lane >> 4;                  // half-wave select

    const int bid = blockIdx.x;
    const int qt  = bid & 127;                   // S/16 tiles
    const int h   = (bid >> 7) & 15;
    const int b   = bid >> 11;

    const long bh    = (long)(b * H_ + h);
    const float* qbase = q + bh * S_ * D_ + (long)qt * 16 * D_;
    const float* kbase = k + bh * S_ * D_;
    const float* vbase = v + bh * S_ * D_;
    const int*   mbase = mask + (long)b * S_ * S_ + (long)qt * 16 * S_;
    float* sbase = score + bh * S_ * S_ + (long)qt * 16 * S_;
    float* obase = out   + bh * S_ * D_ + (long)qt * 16 * D_;

    // ---- Stage 0: cooperative Q-tile load + RoPE -> LDS; zero out_acc ----
    {
        const int row = tid >> 4;                // 0..15
        const int c4  = (tid & 15) << 2;         // 0,4,..,60
        v4f qv = *(const v4f*)(qbase + row * D_ + c4);
        const float t = (float)(qt * 16 + row);
        const int p0 = c4 >> 1;
        float s0, c0, s1, c1;
        __sincosf(t * inv_freq(p0),     &s0, &c0);
        __sincosf(t * inv_freq(p0 + 1), &s1, &c1);
        float* dst = q_lds + row * 68 + c4;
        dst[0] = qv.x * c0 - qv.y * s0;
        dst[1] = qv.y * c0 + qv.x * s0;
        dst[2] = qv.z * c1 - qv.w * s1;
        dst[3] = qv.w * c1 + qv.z * s1;
        for (int i = tid; i < QROWS * 66; i += 256) out_acc[i] = 0.0f;
    }
    __syncthreads();

    // A-matrix fragments of RoPE'd Q (same for all k-tiles of this wave)
    v2f afr[16];
#pragma unroll
    for (int j = 0; j < 16; ++j)
        afr[j] = *(const v2f*)(q_lds + r * 68 + 4 * j + 2 * hi);

    // Per-lane RoPE frequency table: invariant across ALL k-tiles (p = 2j + hi)
    float fr[16];
#pragma unroll
    for (int j = 0; j < 16; ++j)
        fr[j] = inv_freq(2 * j + hi);

    // ---- Stage 1: S = Qr·Krᵀ for this wave's 16 contiguous k-tiles ----
    v8f acc[16];
#pragma unroll
    for (int tt = 0; tt < 16; ++tt) {
        const int   kpos = (w * 16 + tt) * 16 + r;        // key row (B-frag col)
        const float* krow = kbase + (long)kpos * D_;
        const float tpos  = (float)kpos;

        // batch all 16 K fragment loads first so latency overlaps the math
        v2f kv[16];
#pragma unroll
        for (int j = 0; j < 16; ++j)
            kv[j] = *(const v2f*)(krow + 4 * j + 2 * hi);

        v8f c = {};
#pragma unroll
        for (int j = 0; j < 16; ++j) {
            float sn, cs;
            __sincosf(tpos * fr[j], &sn, &cs);
            v2f bfr;
            bfr.x = kv[j].x * cs - kv[j].y * sn;
            bfr.y = kv[j].y * cs + kv[j].x * sn;
            c = __builtin_amdgcn_wmma_f32_16x16x4_f32(false, afr[j], false, bfr,
                                                      (short)0, c, false, false);
        }
        acc[tt] = c;
    }

    // ---- Stage 2: scale + mask ----
#pragma unroll
    for (int tt = 0; tt < 16; ++tt) {
        const int colg = (w * 16 + tt) * 16 + r;
#pragma unroll
        for (int i = 0; i < 8; ++i) {
            const int m = mbase[(i + 8 * hi) * S_ + colg];
            acc[tt][i] = (m == 0) ? -10000.0f : acc[tt][i] * 0.125f;
        }
    }

    // ---- Stage 3: softmax (row max/sum across lanes and waves) ----
    float rmax[8];
#pragma unroll
    for (int i = 0; i < 8; ++i) {
        float m = -3.402823466e38f;
#pragma unroll
        for (int tt = 0; tt < 16; ++tt) m = fmaxf(m, acc[tt][i]);
        rmax[i] = m;
    }
#pragma unroll
    for (int d = 1; d < 16; d <<= 1)
#pragma unroll
        for (int i = 0; i < 8; ++i)
            rmax[i] = fmaxf(rmax[i], __shfl_xor(rmax[i], d, 32));
    if (r == 0) {
#pragma unroll
        for (int i = 0; i < 8; ++i) redmax[w][i + 8 * hi] = rmax[i];
    }
    __syncthreads();
    float gmax[8];
#pragma unroll
    for (int i = 0; i < 8; ++i) {
        float m = -3.402823466e38f;
#pragma unroll
        for (int ww = 0; ww < NWAVES; ++ww) m = fmaxf(m, redmax[ww][i + 8 * hi]);
        gmax[i] = m;
    }

    float rsum[8] = {0, 0, 0, 0, 0, 0, 0, 0};
#pragma unroll
    for (int tt = 0; tt < 16; ++tt)
#pragma unroll
        for (int i = 0; i < 8; ++i) {
            const float e = __expf(acc[tt][i] - gmax[i]);
            acc[tt][i] = e;
            rsum[i] += e;
        }
#pragma unroll
    for (int d = 1; d < 16; d <<= 1)
#pragma unroll
        for (int i = 0; i < 8; ++i)
            rsum[i] += __shfl_xor(rsum[i], d, 32);
    if (r == 0) {
#pragma unroll
        for (int i = 0; i < 8; ++i) redsum[w][i + 8 * hi] = rsum[i];
    }
    __syncthreads();
    float inv[8];
#pragma unroll
    for (int i = 0; i < 8; ++i) {
        float s = 0.0f;
#pragma unroll
        for (int ww = 0; ww < NWAVES; ++ww) s += redsum[ww][i + 8 * hi];
        inv[i] = 1.0f / s;
    }

    // ---- Stage 4: normalize + stream score to global (the 537MB output) ----
#pragma unroll
    for (int tt = 0; tt < 16; ++tt) {
        const int colg = (w * 16 + tt) * 16 + r;
#pragma unroll
        for (int i = 0; i < 8; ++i) {
            const float p = acc[tt][i] * inv[i];
            acc[tt][i] = p;
            sbase[(i + 8 * hi) * S_ + colg] = p;
        }
    }

    // ---- Stage 5: out += P·V over this wave's 256-key segment ----
    // Re-read our own freshly written P segment (same-wave store->load is ordered; L0/L2 hot).
    v8f o[4] = {v8f{}, v8f{}, v8f{}, v8f{}};
    const float* prow = sbase + (long)r * S_;
#pragma unroll 8
    for (int j = 0; j < 64; ++j) {
        const int kk = w * 256 + 4 * j + 2 * hi;
        v2f a = *(const v2f*)(prow + kk);
        const float* vrow = vbase + (long)kk * D_;
#pragma unroll
        for (int n0 = 0; n0 < 4; ++n0) {
            v2f bfr;
            bfr.x = vrow[n0 * 16 + r];
            bfr.y = vrow[D_ + n0 * 16 + r];
            o[n0] = __builtin_amdgcn_wmma_f32_16x16x4_f32(false, a, false, bfr,
                                                          (short)0, o[n0], false, false);
        }
    }

    // Cross-wave reduction of the 16x64 out tile via LDS float atomics
#pragma unroll
    for (int n0 = 0; n0 < 4; ++n0)
#pragma unroll
        for (int i = 0; i < 8; ++i)
            atomicAdd(&out_acc[(i + 8 * hi) * 66 + n0 * 16 + r], o[n0][i]);
    __syncthreads();

    {
        const int idx = tid * 4;                 // 1024 floats / 256 threads
        const int row = idx >> 6;
        const int col = idx & 63;
        const float* src = out_acc + row * 66 + col;
        v4f ov = {src[0], src[1], src[2], src[3]};
        *(v4f*)(obase + row * D_ + col) = ov;
    }
}

extern "C" void kernel_launch(void* const* d_in, const int* in_sizes, int n_in,
                              void* d_out, int out_size, void* d_ws, size_t ws_size,
                              hipStream_t stream) {
    (void)in_sizes; (void)n_in; (void)d_ws; (void)ws_size; (void)out_size;
    const float* q    = (const float*)d_in[0];
    const float* k    = (const float*)d_in[1];
    const float* v    = (const float*)d_in[2];
    const int*   mask = (const int*)d_in[3];
    float* out   = (float*)d_out;                          // [B,H,S,D] first
    float* score = out + (size_t)B_ * H_ * S_ * D_;        // then [B,H,S,S]

    const int blocks = B_ * H_ * (S_ / 16);                // 4096
    rope_attn_fused<<<blocks, 256, 0, stream>>>(q, k, v, mask, out, score);
}